// GraphConvolution_44873818308700
// MI455X (gfx1250) — compile-verified
//
#include <hip/hip_runtime.h>

typedef float v2f __attribute__((ext_vector_type(2)));
typedef float v8f __attribute__((ext_vector_type(8)));

#define N_NODES 100000
#define N_EDGES 1600000
#define IN_F 128
#define OUT_F 128

// LDS x-tile row stride (floats): 128 data + 4 pad -> bank = (m*132 + k) % 64
// varies with m (132 % 64 == 4), so the 16-lane A fetch is conflict-free.
#define XS_STRIDE 132

// Guaranteed-native, non-returning f32 atomic add at device scope.
// RMW executes at L2 (out[] is L2-resident: 51.2 MB < 192 MB), tracked by STOREcnt.
__device__ __forceinline__ void atomic_fadd_f32(float* p, float v) {
    asm volatile("global_atomic_add_f32 %0, %1, off scope:SCOPE_DEV"
                 :: "v"(p), "v"(v)
                 : "memory");
}

// ---------------------------------------------------------------------------
// Kernel 3 (listed first for disasm visibility): COO scatter.
// One wave per edge: 32 lanes x float4 = 128 features. Gather from support
// (L2-resident, contiguous 512B per wave), scale by edge_val, then 4
// non-returning f32 atomic adds per lane into out[row].
// ---------------------------------------------------------------------------
__global__ __launch_bounds__(256)
void gcn_edge_scatter_kernel(const int* __restrict__ erow,
                             const int* __restrict__ ecol,
                             const float* __restrict__ eval,
                             const float* __restrict__ support,
                             float* __restrict__ out) {
    const int e = blockIdx.x * 8 + (threadIdx.x >> 5);
    if (e >= N_EDGES) return;
    const int lane = threadIdx.x & 31;

    const int   row = erow[e];
    const int   col = ecol[e];
    const float val = eval[e];

    const float4 s = *(const float4*)(support + (size_t)col * OUT_F + lane * 4);
    float* dst = out + (size_t)row * OUT_F + lane * 4;

    atomic_fadd_f32(dst + 0, val * s.x);
    atomic_fadd_f32(dst + 1, val * s.y);
    atomic_fadd_f32(dst + 2, val * s.z);
    atomic_fadd_f32(dst + 3, val * s.w);
}

// ---------------------------------------------------------------------------
// Kernel 2: out[n][f] = b[f]   (bias init; atomics accumulate on top)
// ---------------------------------------------------------------------------
__global__ __launch_bounds__(256)
void gcn_bias_init_kernel(const float* __restrict__ b, float* __restrict__ out) {
    int i = blockIdx.x * 256 + threadIdx.x;          // N*OUT_F = 12.8M < 2^31
    if (i < N_NODES * OUT_F) out[i] = b[i & (OUT_F - 1)];
}

// ---------------------------------------------------------------------------
// Kernel 1: support[N,128] = x[N,128] @ weight[128,128] via v_wmma_f32_16x16x4_f32
// Block = 256 threads = 8 waves. Block tile: 16 rows x 128 cols.
// Wave w computes the 16x16 tile at cols [16w, 16w+16), K-loop 128 in steps of 4.
// ---------------------------------------------------------------------------
__global__ __launch_bounds__(256)
void gcn_gemm_wmma_kernel(const float* __restrict__ x,
                          const float* __restrict__ w,
                          float* __restrict__ support) {
    __shared__ float xs[16 * XS_STRIDE];   // ~8.3 KB

    const int rowBase = blockIdx.x * 16;

    // Cooperative float4 staging of the 16x128 x-tile (512 float4 / 256 thr = 2 each).
    {
        const float4* xg = (const float4*)(x + (size_t)rowBase * IN_F);
        float4* xs4 = (float4*)xs;
        #pragma unroll
        for (int i = 0; i < 2; ++i) {
            int idx = threadIdx.x + i * 256;   // 0..511
            int r = idx >> 5;                  // row 0..15
            int q = idx & 31;                  // float4 index within row
            xs4[r * (XS_STRIDE / 4) + q] = xg[r * (IN_F / 4) + q];
        }
    }
    __syncthreads();

    const int wave = threadIdx.x >> 5;
    const int lane = threadIdx.x & 31;
    const int lmod = lane & 15;   // M (for A) / N (for B, D)
    const int lhi  = lane >> 4;   // selects K pair (A/B) / +8 rows (D)
    const int colBase = wave * 16;

    v8f acc = {};
    #pragma unroll 4
    for (int k = 0; k < IN_F; k += 4) {
        const int ka = k + 2 * lhi;           // even -> 8B aligned LDS read
        // A (16x4): VGPR r holds x[rowBase + lmod][ka + r]
        v2f a = *(const v2f*)(xs + lmod * XS_STRIDE + ka);
        // B (4x16): VGPR r holds w[ka + r][colBase + lmod]
        v2f b;
        b.x = w[(ka    ) * OUT_F + colBase + lmod];
        b.y = w[(ka + 1) * OUT_F + colBase + lmod];
        // 8 args: (neg_a, A, neg_b, B, c_mod, C, reuse_a, reuse_b)
        acc = __builtin_amdgcn_wmma_f32_16x16x4_f32(
            false, a, false, b, (short)0, acc, false, false);
    }

    // D layout: VGPR r, lane -> row = rowBase + r + 8*lhi, col = colBase + lmod
    float* outp = support + (size_t)(rowBase + 8 * lhi) * OUT_F + colBase + lmod;
    #pragma unroll
    for (int r = 0; r < 8; ++r) {
        outp[(size_t)r * OUT_F] = acc[r];
    }
}

// ---------------------------------------------------------------------------
extern "C" void kernel_launch(void* const* d_in, const int* in_sizes, int n_in,
                              void* d_out, int out_size, void* d_ws, size_t ws_size,
                              hipStream_t stream) {
    const float* x      = (const float*)d_in[0];   // [N, 128]
    const int*   erow   = (const int*)  d_in[1];   // [E]
    const int*   ecol   = (const int*)  d_in[2];   // [E]
    const float* eval   = (const float*)d_in[3];   // [E]
    const float* weight = (const float*)d_in[4];   // [128, 128]
    const float* bias   = (const float*)d_in[5];   // [128]
    float* out     = (float*)d_out;                // [N, 128]
    float* support = (float*)d_ws;                 // needs N*128*4 = 51.2 MB scratch

    // 1) dense projection (WMMA f32)
    gcn_gemm_wmma_kernel<<<N_NODES / 16, 256, 0, stream>>>(x, weight, support);

    // 2) out = bias (broadcast)
    gcn_bias_init_kernel<<<(N_NODES * OUT_F + 255) / 256, 256, 0, stream>>>(bias, out);

    // 3) edge scatter-accumulate (same stream -> ordered after 1 and 2)
    gcn_edge_scatter_kernel<<<(N_EDGES + 7) / 8, 256, 0, stream>>>(
        erow, ecol, eval, support, out);
}